// LocationSensitiveAttention_65223373357360
// MI455X (gfx1250) — compile-verified
//
#include <hip/hip_runtime.h>
#include <stdint.h>
#include <math.h>

#define T_DIM 4096
#define B_DIM 64
#define H_DIM 128
#define Q_DIM 1024
#define K_CONV 31
#define PAD_L 15
#define WL 128

typedef float v2f __attribute__((ext_vector_type(2)));
typedef float v8f __attribute__((ext_vector_type(8)));

// ---------------------------------------------------------------------------
// Kernel 1: qp[b,h] = sum_q query[b,q] * Wq[h,q] + bq[h]
// M=64 (4 tiles) x N=128 (8 tiles) x K=1024, via V_WMMA_F32_16X16X4_F32.
// A lane layout (16x4 f32): lanes 0-15 -> M=m0+lane, K={0,1}; lanes 16-31 ->
// M=m0+lane-16, K={2,3}. B mirrors with N striped across lanes.
// ---------------------------------------------------------------------------
__global__ __launch_bounds__(256) void qp_gemm_wmma(
    const float* __restrict__ query,  // [1,B,Q] -> [B,Q]
    const float* __restrict__ Wq,     // [H,Q]
    const float* __restrict__ bq,     // [H]
    float* __restrict__ qp)           // [B,H]
{
  const int wave = (blockIdx.x * blockDim.x + threadIdx.x) >> 5;  // 0..31
  const int lane = threadIdx.x & 31;
  const int m0 = (wave >> 3) * 16;   // 4 M tiles
  const int n0 = (wave & 7) * 16;    // 8 N tiles

  const int half = lane >> 4;        // 0: K pair {0,1}, 1: K pair {2,3}
  const int l15  = lane & 15;

  const float* arow = query + (size_t)(m0 + l15) * Q_DIM + half * 2;
  const float* brow = Wq    + (size_t)(n0 + l15) * Q_DIM + half * 2;  // B[k][n]=Wq[n*Q+k]

  v8f acc = {};
  for (int k = 0; k < Q_DIM; k += 4) {
    v2f a = *(const v2f*)(arow + k);
    v2f b = *(const v2f*)(brow + k);
    acc = __builtin_amdgcn_wmma_f32_16x16x4_f32(
        /*neg_a=*/false, a, /*neg_b=*/false, b,
        /*c_mod=*/(short)0, acc, /*reuse_a=*/false, /*reuse_b=*/false);
  }

  // D layout: VGPR r -> M = m0 + r + 8*half, N = n0 + l15
  const int n = n0 + l15;
  const float bias = bq[n];
#pragma unroll
  for (int r = 0; r < 8; ++r) {
    const int m = m0 + r + half * 8;
    qp[m * H_DIM + n] = acc[r] + bias;
  }
}

// ---------------------------------------------------------------------------
// Kernel 2: per-batch windowed attention. One block (256 thr) per batch.
// conv(31-tap) + tanh + v-dot -> masked softmax -> context, argmax, ws_new.
// ---------------------------------------------------------------------------
__global__ __launch_bounds__(256) void attn_window(
    const float* __restrict__ enc,        // [T,B,H]
    const uint8_t* __restrict__ mask,     // [B,T] (bool, 1 byte)
    const int* __restrict__ num_tokens,   // [B]
    const float* __restrict__ cum,        // [B,T]
    const float* __restrict__ init_cum,   // [B,1]
    const int* __restrict__ wstart,       // [B]
    const float* __restrict__ conv_w,     // [H,1,K]
    const float* __restrict__ conv_b,     // [H]
    const float* __restrict__ vvec,       // [H]
    const float* __restrict__ qp,         // [B,H]
    float* __restrict__ align_w_out,      // ws: [B,WL]
    float* __restrict__ context_out,      // d_out: [B,H]
    float* __restrict__ wsnew_out)        // d_out: [B]
{
  __shared__ float s_loc[WL + 2 * PAD_L];     // 158
  __shared__ float s_cw[H_DIM * K_CONV];      // 3968
  __shared__ float s_bias[H_DIM];
  __shared__ float s_v[H_DIM];
  __shared__ float s_score[WL];
  __shared__ float s_align[WL];
  __shared__ float s_red[256];
  __shared__ int   s_idx[256];

  const int b   = blockIdx.x;
  const int tid = threadIdx.x;
  const int ws  = wstart[b];
  const float ic = init_cum[b];

  // Stage padded alignment window: loc[gi] = ic (gi<PAD) | masked cum | 0
  for (int j = tid; j < WL + 2 * PAD_L; j += 256) {
    const int gi = ws + j;                 // gi in [0, T + 2*PAD)
    float val = 0.f;
    if (gi < PAD_L) {
      val = ic;
    } else {
      const int t = gi - PAD_L;
      if (t < T_DIM) val = mask[b * T_DIM + t] ? cum[b * T_DIM + t] : 0.f;
    }
    s_loc[j] = val;
  }
  for (int i = tid; i < H_DIM * K_CONV; i += 256) s_cw[i] = conv_w[i];
  if (tid < H_DIM) {
    s_bias[tid] = conv_b[tid] + qp[b * H_DIM + tid];
    s_v[tid]    = vvec[tid];
  }
  if (tid < WL) s_score[tid] = 0.f;
  __syncthreads();

  // conv + tanh + dot with v; thread t handles w = t&127, h strided by 2
  {
    const int w  = tid & (WL - 1);
    const int h0 = tid >> 7;  // 0 or 1
    float local = 0.f;
    for (int h = h0; h < H_DIM; h += 2) {
      float acc = s_bias[h];
      const float* cw = &s_cw[h * K_CONV];
#pragma unroll
      for (int k = 0; k < K_CONV; ++k) acc += cw[k] * s_loc[w + k];
      local += s_v[h] * tanhf(acc);
    }
    atomicAdd(&s_score[w], local);   // ds_add_f32
  }
  __syncthreads();

  // masked softmax over window
  float msc = -INFINITY;
  if (tid < WL) {
    const bool mv = mask[b * T_DIM + ws + tid] != 0;
    msc = mv ? s_score[tid] : -INFINITY;
  }
  s_red[tid] = (tid < WL) ? msc : -INFINITY;
  __syncthreads();
  for (int st = 128; st > 0; st >>= 1) {
    if (tid < st) s_red[tid] = fmaxf(s_red[tid], s_red[tid + st]);
    __syncthreads();
  }
  const float smax = s_red[0];
  __syncthreads();

  float e = 0.f;
  if (tid < WL) e = expf(msc - smax);   // masked -> exp(-inf)=0
  s_red[tid] = (tid < WL) ? e : 0.f;
  __syncthreads();
  for (int st = 128; st > 0; st >>= 1) {
    if (tid < st) s_red[tid] += s_red[tid + st];
    __syncthreads();
  }
  const float ssum = s_red[0];
  __syncthreads();

  float al = 0.f;
  if (tid < WL) {
    al = e / ssum;
    s_align[tid] = al;
    align_w_out[b * WL + tid] = al;
  }
  __syncthreads();

  // context[b,h] = sum_w align[w] * enc[ws+w, b, h]  (coalesced along h)
  if (tid < H_DIM) {
    float acc = 0.f;
    const float* ep = enc + (size_t)ws * B_DIM * H_DIM + (size_t)b * H_DIM + tid;
    for (int w2 = 0; w2 < WL; ++w2)
      acc += s_align[w2] * ep[(size_t)w2 * B_DIM * H_DIM];
    context_out[b * H_DIM + tid] = acc;
  }

  // first-index argmax over align (max > 0, so window dominates align_full)
  s_red[tid] = (tid < WL) ? al : -INFINITY;
  s_idx[tid] = (tid < WL) ? tid : WL;
  __syncthreads();
  for (int st = 128; st > 0; st >>= 1) {
    if (tid < st) {
      const float v2 = s_red[tid + st];
      const int   i2 = s_idx[tid + st];
      if (v2 > s_red[tid] || (v2 == s_red[tid] && i2 < s_idx[tid])) {
        s_red[tid] = v2;
        s_idx[tid] = i2;
      }
    }
    __syncthreads();
  }
  if (tid == 0) {
    int wn = ws + s_idx[0] - WL / 2;
    const int hi = num_tokens[b] - WL;
    wn = wn < hi ? wn : hi;
    wn = wn > 0 ? wn : 0;
    wsnew_out[b] = (float)wn;   // int result stored into float-typed d_out
  }
}

// ---------------------------------------------------------------------------
// Kernel 3: full-length scatter: align_full and cum_new over [B,T].
// ---------------------------------------------------------------------------
__global__ __launch_bounds__(256) void scatter_full(
    const uint8_t* __restrict__ mask, const float* __restrict__ cum,
    const int* __restrict__ wstart, const float* __restrict__ align_w,
    float* __restrict__ cum_new, float* __restrict__ align_full)
{
  const int i = blockIdx.x * 256 + threadIdx.x;
  if (i >= B_DIM * T_DIM) return;
  const int b = i >> 12;            // T = 4096
  const int t = i & (T_DIM - 1);
  const float c = mask[i] ? cum[i] : 0.f;
  const int j = t - wstart[b];
  const float a = (j >= 0 && j < WL) ? align_w[b * WL + j] : 0.f;
  align_full[i] = a;
  cum_new[i]   = c + a;
}

// ---------------------------------------------------------------------------
extern "C" void kernel_launch(void* const* d_in, const int* in_sizes, int n_in,
                              void* d_out, int out_size, void* d_ws, size_t ws_size,
                              hipStream_t stream) {
  const float*   enc      = (const float*)d_in[0];   // [T,B,H]
  const uint8_t* mask     = (const uint8_t*)d_in[1]; // [B,T] bool
  const int*     ntok     = (const int*)d_in[2];     // [B]
  const float*   query    = (const float*)d_in[3];   // [1,B,Q]
  const float*   cum      = (const float*)d_in[4];   // [B,T]
  const float*   init_cum = (const float*)d_in[5];   // [B,1]
  const int*     wstart   = (const int*)d_in[6];     // [B]
  const float*   Wq       = (const float*)d_in[7];   // [H,Q]
  const float*   bq       = (const float*)d_in[8];   // [H]
  const float*   conv_w   = (const float*)d_in[9];   // [H,1,K]
  const float*   conv_b   = (const float*)d_in[10];  // [H]
  const float*   vvec     = (const float*)d_in[11];  // [H]

  float* out = (float*)d_out;
  float* context_out = out;                                   // [B,H]   8192
  float* cum_new     = out + B_DIM * H_DIM;                   // [B,T] 262144
  float* align_full  = cum_new + B_DIM * T_DIM;               // [B,T] 262144
  float* wsnew_out   = align_full + B_DIM * T_DIM;            // [B]       64

  float* qp      = (float*)d_ws;                              // [B,H]
  float* align_w = qp + B_DIM * H_DIM;                        // [B,WL]

  // 1) query projection via f32 WMMA: 32 waves = 4 blocks x 8 waves
  qp_gemm_wmma<<<4, 256, 0, stream>>>(query, Wq, bq, qp);

  // 2) per-batch windowed attention
  attn_window<<<B_DIM, 256, 0, stream>>>(enc, mask, ntok, cum, init_cum, wstart,
                                         conv_w, conv_b, vvec, qp,
                                         align_w, context_out, wsnew_out);

  // 3) full-length scatter of alignment + cumulative update
  scatter_full<<<(B_DIM * T_DIM + 255) / 256, 256, 0, stream>>>(
      mask, cum, wstart, align_w, cum_new, align_full);
}